// AudioClassifier_86208583565452
// MI455X (gfx1250) — compile-verified
//
#include <hip/hip_runtime.h>

// SNN audio classifier, MI455X (gfx1250, wave32).
// One workgroup (128 threads = 4 waves) per batch element; sequential T loop
// with all LIF membrane state in LDS.
//  - conv1: fp32 WMMA 16x16x4 GEMM, M=16 ch x N=136 pos (9 tiles over 4 waves) x K=13->16
//  - conv2: fp32 WMMA 16x16x4 GEMM, M=32 ch x N=22 pos x K=112, one 16x16 tile per wave
//  - fc1/fc2: VALU matvec (M=1 per batch element -> WMMA would be 1/16 efficient)

typedef __attribute__((ext_vector_type(2))) float v2f;
typedef __attribute__((ext_vector_type(8))) float v8f;

#define T_STEPS 100
#define L_IN    686
#define XPAD    736     // padded conv1 input (1 left pad + 686 + zero tail for ghost cols)
#define C1      16
#define KW1     13
#define L1      136     // (686 + 2 - 13)/5 + 1
#define LP      68      // after maxpool2
#define C2      32
#define KW2     7
#define L2      22      // (68 + 2 - 7)/3 + 1
#define KTOT    112     // C1 * KW2 (conv2 GEMM K)
#define PW      104     // pool row pitch (>= 3*31+6+1, zero padded)
#define F1      704     // C2 * L2
#define H1      32
#define BETA    0.9f
#define THETA   1.0f

__global__ __launch_bounds__(128, 1)
void snn_lif_scan_kernel(const float* __restrict__ x,
                         const float* __restrict__ w1,  const float* __restrict__ b1,
                         const float* __restrict__ w2,  const float* __restrict__ b2,
                         const float* __restrict__ wf1, const float* __restrict__ bf1,
                         const float* __restrict__ wf2, const float* __restrict__ bf2,
                         float* __restrict__ out)
{
    __shared__ float s_xp[XPAD];            // conv1 input, zero padded
    __shared__ float s_mem1[C1 * L1];       // LIF1 state
    __shared__ float s_spk1[C1 * L1];
    __shared__ float s_pool[C1 * PW];       // pooled spikes, zero padded rows
    __shared__ float s_mem2[C2 * L2];       // LIF2 state
    __shared__ float s_spk2[F1];            // fc1 input (flat [c2*22+p])
    __shared__ float s_part[128];           // fc1 partial sums
    __shared__ float s_mem3[H1];
    __shared__ float s_spk3[H1];
    __shared__ float s_mem4[2];
    __shared__ float s_acc[2];

    const int tid  = threadIdx.x;
    const int b    = blockIdx.x;
    const int lane = tid & 31;
    const int wave = tid >> 5;
    const int lg   = (lane >> 4) & 1;   // lane group (0: lanes 0-15, 1: 16-31)
    const int ln16 = lane & 15;
    const int mt   = wave & 1;          // conv2 M tile: out-channel group
    const int nt   = wave >> 1;         // conv2 N tile: position group

    // ---- one-time init of persistent LDS state ----
    for (int i = tid; i < XPAD; i += 128)    s_xp[i]   = 0.f;  // pads/tail stay 0 forever
    for (int i = tid; i < C1 * L1; i += 128) s_mem1[i] = 0.f;
    for (int i = tid; i < C1 * PW; i += 128) s_pool[i] = 0.f;  // pads stay 0 forever
    for (int i = tid; i < C2 * L2; i += 128) s_mem2[i] = 0.f;
    if (tid < H1) s_mem3[tid] = 0.f;
    if (tid < 2) { s_mem4[tid] = 0.f; s_acc[tid] = 0.f; }

    // ---- conv1 WMMA A fragments (w1, K padded 13->16), identical in all waves ----
    // A layout (16x4 f32): lane holds A[m = lane%16][k = 4kc + 2*lg + {0,1}]
    v2f a1frag[4];
    #pragma unroll
    for (int kc = 0; kc < 4; ++kc) {
        int k0 = 4 * kc + 2 * lg;
        a1frag[kc].x = (k0     < KW1) ? w1[ln16 * KW1 + k0]     : 0.f;
        a1frag[kc].y = (k0 + 1 < KW1) ? w1[ln16 * KW1 + k0 + 1] : 0.f;
    }
    float bias1[8];
    #pragma unroll
    for (int g = 0; g < 8; ++g) bias1[g] = b1[8 * lg + g];

    // ---- conv2 WMMA A fragments (w2 as [32 x 112]) ----
    const int ch_a = 16 * mt + ln16;
    v2f afrag[28];
    #pragma unroll
    for (int kc = 0; kc < 28; ++kc) {
        int k0 = 4 * kc + 2 * lg;
        afrag[kc].x = w2[ch_a * KTOT + k0];
        afrag[kc].y = w2[ch_a * KTOT + k0 + 1];
    }
    // C/D layout: vgpr g, lane -> row ch = 16*mt + 8*lg + g, col p = 16*nt + ln16
    float bias2[8];
    #pragma unroll
    for (int g = 0; g < 8; ++g) bias2[g] = b2[16 * mt + 8 * lg + g];

    const float bf1_j = bf1[tid & 31];
    const float* __restrict__ xb = x + (size_t)b * T_STEPS * L_IN;
    const int p_b = 16 * nt + ln16;     // conv2 lane column (output position)

    __syncthreads();

    for (int t = 0; t < T_STEPS; ++t) {
        // ---- stage x_t (padded) + prefetch next timestep's slice ----
        for (int i = tid; i < L_IN; i += 128) s_xp[1 + i] = xb[t * L_IN + i];
        if (t + 1 < T_STEPS && tid < 22)                  // 22 x 128B lines cover 686 floats
            __builtin_prefetch(xb + (t + 1) * L_IN + tid * 32, 0, 1);
        __syncthreads();

        // ---- conv1 as fp32 WMMA GEMM: D[16ch x 136pos] = W1[16x16] * im2col[16 x 136] ----
        // 9 N-tiles distributed over the 4 waves (wave-uniform trip counts)
        for (int nt2 = wave; nt2 < 9; nt2 += 4) {
            const int p1 = 16 * nt2 + ln16;               // 0..143 (>=136 are ghosts)
            v8f acc1 = {0.f, 0.f, 0.f, 0.f, 0.f, 0.f, 0.f, 0.f};
            #pragma unroll
            for (int kc = 0; kc < 4; ++kc) {
                int k0 = 4 * kc + 2 * lg;
                // B[k][n] = x_pad[5*p - 1 + k]  (padded index = 5*p + k); zero for k>=13
                v2f bfr;
                bfr.x = (k0     < KW1) ? s_xp[5 * p1 + k0]     : 0.f;
                bfr.y = (k0 + 1 < KW1) ? s_xp[5 * p1 + k0 + 1] : 0.f;
                acc1 = __builtin_amdgcn_wmma_f32_16x16x4_f32(
                    false, a1frag[kc], false, bfr, (short)0, acc1, false, false);
            }
            // LIF1 directly on accumulator registers
            #pragma unroll
            for (int g = 0; g < 8; ++g) {
                int c = 8 * lg + g;
                if (p1 < L1) {
                    int idx     = c * L1 + p1;
                    float cur   = acc1[g] + bias1[g];
                    float mp    = s_mem1[idx];
                    float reset = (mp > THETA) ? THETA : 0.f;   // reset from PREVIOUS mem
                    float m     = fmaf(BETA, mp, cur) - reset;
                    s_mem1[idx] = m;
                    s_spk1[idx] = (m > THETA) ? 1.f : 0.f;
                }
            }
        }
        __syncthreads();

        // ---- maxpool2 into padded pool rows ----
        for (int idx = tid; idx < C1 * LP; idx += 128) {
            int c = idx / LP, lp = idx - c * LP;
            float a0 = s_spk1[c * L1 + 2 * lp];
            float a1 = s_spk1[c * L1 + 2 * lp + 1];
            s_pool[c * PW + 1 + lp] = fmaxf(a0, a1);
        }
        __syncthreads();

        // ---- conv2 as fp32 WMMA GEMM: D[32ch x 22pos] = W2[32x112] * im2col[112 x 22] ----
        v8f acc = {0.f, 0.f, 0.f, 0.f, 0.f, 0.f, 0.f, 0.f};
        #pragma unroll
        for (int kc = 0; kc < 28; ++kc) {
            int k0 = 4 * kc + 2 * lg;
            int k1 = k0 + 1;
            // B[k][n] = pooled[c1 = k/7][3*p - 1 + (k%7)]  (padded index = 3*p + k%7)
            v2f bfr;
            bfr.x = s_pool[(k0 / 7) * PW + 3 * p_b + (k0 % 7)];
            bfr.y = s_pool[(k1 / 7) * PW + 3 * p_b + (k1 % 7)];
            acc = __builtin_amdgcn_wmma_f32_16x16x4_f32(
                false, afrag[kc], false, bfr, (short)0, acc, false, false);
        }
        // LIF2 directly on accumulator registers
        #pragma unroll
        for (int g = 0; g < 8; ++g) {
            int ch = 16 * mt + 8 * lg + g;
            if (p_b < L2) {
                int idx     = ch * L2 + p_b;
                float cur   = acc[g] + bias2[g];
                float mp    = s_mem2[idx];
                float reset = (mp > THETA) ? THETA : 0.f;
                float m     = fmaf(BETA, mp, cur) - reset;
                s_mem2[idx] = m;
                s_spk2[idx] = (m > THETA) ? 1.f : 0.f;
            }
        }
        __syncthreads();

        // ---- fc1: 32 x 704 matvec, 4 partial lanes per output ----
        {
            int j = tid & 31, part = tid >> 5;
            const float4* w4 = (const float4*)(wf1 + j * F1 + part * 176);
            const float4* s4 = (const float4*)(s_spk2 + part * 176);
            float s = 0.f;
            #pragma unroll 4
            for (int i = 0; i < 44; ++i) {
                float4 wv = w4[i], sv = s4[i];
                s = fmaf(wv.x, sv.x, s); s = fmaf(wv.y, sv.y, s);
                s = fmaf(wv.z, sv.z, s); s = fmaf(wv.w, sv.w, s);
            }
            s_part[tid] = s;
        }
        __syncthreads();
        if (tid < H1) {
            float h = bf1_j + s_part[tid] + s_part[tid + 32] + s_part[tid + 64] + s_part[tid + 96];
            float mp    = s_mem3[tid];
            float reset = (mp > THETA) ? THETA : 0.f;
            float m     = fmaf(BETA, mp, h) - reset;
            s_mem3[tid] = m;
            s_spk3[tid] = (m > THETA) ? 1.f : 0.f;
        }
        __syncthreads();

        // ---- fc2 + LIF4 + spike accumulation ----
        if (tid < 2) {
            float h = bf2[tid];
            #pragma unroll
            for (int j = 0; j < 32; ++j) h = fmaf(wf2[tid * 32 + j], s_spk3[j], h);
            float mp    = s_mem4[tid];
            float reset = (mp > THETA) ? THETA : 0.f;
            float m     = fmaf(BETA, mp, h) - reset;
            s_mem4[tid] = m;
            s_acc[tid] += (m > THETA) ? 1.f : 0.f;
        }
        __syncthreads();
    }

    if (tid < 2) out[b * 2 + tid] = s_acc[tid];
}

extern "C" void kernel_launch(void* const* d_in, const int* in_sizes, int n_in,
                              void* d_out, int out_size, void* d_ws, size_t ws_size,
                              hipStream_t stream) {
    (void)n_in; (void)out_size; (void)d_ws; (void)ws_size;
    const float* x   = (const float*)d_in[0];
    const float* w1  = (const float*)d_in[1];
    const float* b1  = (const float*)d_in[2];
    const float* w2  = (const float*)d_in[3];
    const float* b2  = (const float*)d_in[4];
    const float* wf1 = (const float*)d_in[5];
    const float* bf1 = (const float*)d_in[6];
    const float* wf2 = (const float*)d_in[7];
    const float* bf2 = (const float*)d_in[8];

    const int B = in_sizes[0] / (T_STEPS * L_IN);   // 512
    snn_lif_scan_kernel<<<B, 128, 0, stream>>>(x, w1, b1, w2, b2, wf1, bf1, wf2, bf2,
                                               (float*)d_out);
}